// TopoFeatModel_58282706207194
// MI455X (gfx1250) — compile-verified
//
#include <hip/hip_runtime.h>
#include <hip/hip_bf16.h>

// ---------------------------------------------------------------------------
// 2-layer GCN (PyG GCNConv semantics) for MI455X / gfx1250.
// N=100000 nodes, E=6.4M edges, F: 8 -> 16 -> 16.
//
// Roofline: dense matmuls are ~77 MFLOP total (negligible) -> done with
// V_WMMA_F32_16X16X4_F32 at full f32 precision.  The edge gather/scatter
// dominates: node features (~20 MB) are L2-resident (192 MB L2), so gathers
// and the 16 f32 scatter-adds/edge resolve in L2 at atomic throughput, and
// HBM traffic reduces to streaming the edge list (~128 MB/layer, NT-hinted
// so it never evicts the feature working set).
// ---------------------------------------------------------------------------

typedef float v2f __attribute__((ext_vector_type(2)));
typedef float v8f __attribute__((ext_vector_type(8)));

#define FOUT 16

// Native HW f32 atomic add (global_atomic_add_f32, no CAS loop).
__device__ __forceinline__ void atom_add_f32(float* p, float v) {
    unsafeAtomicAdd(p, v);
}

// ---------------- small utility kernels ----------------

__global__ void fill_f32(float* __restrict__ p, float v, int n) {
    int i = blockIdx.x * blockDim.x + threadIdx.x;
    if (i < n) p[i] = v;
}

// deg[col] += w  (self-loop weight 1.0 pre-filled by fill_f32)
__global__ void accum_deg(const long long* __restrict__ ei,   // [2,E] flat
                          const float* __restrict__ w,
                          float* __restrict__ deg, int E) {
    int e = blockIdx.x * blockDim.x + threadIdx.x;
    if (e >= E) return;
    long long c = __builtin_nontemporal_load(ei + (size_t)E + e);
    float     x = __builtin_nontemporal_load(w + e);
    atom_add_f32(&deg[c], x);
}

// deg -> dinv in place
__global__ void deg_to_dinv(float* __restrict__ deg, int n) {
    int i = blockIdx.x * blockDim.x + threadIdx.x;
    if (i >= n) return;
    float d = deg[i];
    deg[i] = (d > 0.0f) ? rsqrtf(d) : 0.0f;
}

// ---------------- WMMA dense matmul: out[N,16] = in[N,K] @ W[K,16] ----------
// One wave32 per 16-row output tile; K is compile-time (8 or 16) so the body
// is straight-line: K/4 back-to-back v_wmma_f32_16x16x4_f32.
// A (16x4 f32): lanes 0-15 -> K=k0,k0+1 ; lanes 16-31 -> K=k0+2,k0+3 (ISA 7.12.2)
// B (4x16 f32): mirrored, N = lane&15.
// C/D (16x16 f32): VGPR v -> row v (lanes 0-15) / v+8 (lanes 16-31), N = lane&15.

template <int K>
__global__ __launch_bounds__(256) void gcn_matmul_wmma(
    const float* __restrict__ in, const float* __restrict__ W,
    float* __restrict__ out, int ntiles)
{
    const int lane = threadIdx.x & 31;
    const int wave = threadIdx.x >> 5;
    const int tile = blockIdx.x * 8 + wave;
    if (tile >= ntiles) return;           // wave-uniform: EXEC stays all-ones

    const int  m    = lane & 15;          // row within tile (A) / column N (B,C)
    const int  half = lane >> 4;          // 0: lanes 0-15, 1: lanes 16-31
    const long row  = (long)tile * 16 + m;

    v8f c = {};
#pragma unroll
    for (int k0 = 0; k0 < K; k0 += 4) {
        const int kb = k0 + half * 2;
        v2f a, b;
        a.x = in[row * K + kb];
        a.y = in[row * K + kb + 1];
        b.x = W[(long)kb       * FOUT + m];
        b.y = W[(long)(kb + 1) * FOUT + m];
        // (neg_a, A, neg_b, B, c_mod, C, reuse_a, reuse_b)
        c = __builtin_amdgcn_wmma_f32_16x16x4_f32(
                false, a, false, b, (short)0, c, false, false);
    }

#pragma unroll
    for (int v = 0; v < 8; ++v) {
        const int rM = v + half * 8;
        out[((long)tile * 16 + rM) * FOUT + m] = c[v];
    }
}

// ---------------- edge gather-scale-scatter ----------------
// One thread per edge: acc[col] += h[row] * (dinv[row]*w*dinv[col]).
// Edge stream: NT loads (read once, never re-used).  Feature gathers: 4x b128
// loads hitting L2.  Scatter: 16 native f32 atomics (no return -> STOREcnt).

__global__ __launch_bounds__(256) void gcn_edge_scatter(
    const float* __restrict__ h,
    const long long* __restrict__ ei,      // [2,E] flat
    const float* __restrict__ w,
    const float* __restrict__ dinv,
    float* __restrict__ acc, int E)
{
    int e = blockIdx.x * blockDim.x + threadIdx.x;
    if (e >= E) return;
    long long r  = __builtin_nontemporal_load(ei + e);
    long long cc = __builtin_nontemporal_load(ei + (size_t)E + e);
    float     we = __builtin_nontemporal_load(w + e);
    float norm = dinv[r] * we * dinv[cc];

    const float4* hr = (const float4*)(h + (size_t)r * FOUT);
    float* ac = acc + (size_t)cc * FOUT;
#pragma unroll
    for (int q = 0; q < 4; ++q) {
        float4 v = hr[q];
        atom_add_f32(ac + q * 4 + 0, v.x * norm);
        atom_add_f32(ac + q * 4 + 1, v.y * norm);
        atom_add_f32(ac + q * 4 + 2, v.z * norm);
        atom_add_f32(ac + q * 4 + 3, v.w * norm);
    }
}

// ---------------- finalize: self-loop + bias (+ optional ReLU), in place OK --

__global__ void gcn_finalize(const float* __restrict__ acc,
                             const float* __restrict__ hself,
                             const float* __restrict__ dinv,
                             const float* __restrict__ bias,
                             float* __restrict__ out,
                             int total, int do_relu)
{
    int i = blockIdx.x * blockDim.x + threadIdx.x;
    if (i >= total) return;
    int node = i >> 4;
    int f    = i & 15;
    float d  = dinv[node];
    float v  = acc[i] + hself[i] * d * d + bias[f];
    if (do_relu) v = fmaxf(v, 0.0f);
    out[i] = v;
}

// ---------------------------------------------------------------------------

extern "C" void kernel_launch(void* const* d_in, const int* in_sizes, int n_in,
                              void* d_out, int out_size, void* d_ws, size_t ws_size,
                              hipStream_t stream) {
    const float*     x  = (const float*)d_in[0];
    const long long* ei = (const long long*)d_in[1];   // int64 [2,E]
    const float*     w  = (const float*)d_in[2];
    const float*     W1 = (const float*)d_in[3];
    const float*     b1 = (const float*)d_in[4];
    const float*     W2 = (const float*)d_in[5];
    const float*     b2 = (const float*)d_in[6];

    const int FIN = 8;
    const int N = in_sizes[0] / FIN;      // 100000
    const int E = in_sizes[2];            // 6400000
    const int NF = N * FOUT;

    // workspace layout
    float* dinv = (float*)d_ws;           // [N]     deg -> dinv in place
    float* h1   = dinv + N;               // [N,16]  x@W1
    float* acc1 = h1 + NF;                // [N,16]  conv1 accum -> relu output
    float* h2   = acc1 + NF;              // [N,16]  relu(..)@W2
    float* outf = (float*)d_out;          // [N,16]  conv2 accum -> final

    const int B = 256;
    const int ntiles = (N + 15) / 16;     // 6250 exactly
    dim3 mmGrid((ntiles + 7) / 8);

    // ---- normalization (shared by both layers) ----
    fill_f32<<<(N + B - 1) / B, B, 0, stream>>>(dinv, 1.0f, N);           // self-loop wt
    accum_deg<<<(E + B - 1) / B, B, 0, stream>>>(ei, w, dinv, E);
    deg_to_dinv<<<(N + B - 1) / B, B, 0, stream>>>(dinv, N);

    // ---- layer 1 ----
    gcn_matmul_wmma<8><<<mmGrid, B, 0, stream>>>(x, W1, h1, ntiles);
    fill_f32<<<(NF + B - 1) / B, B, 0, stream>>>(acc1, 0.0f, NF);
    gcn_edge_scatter<<<(E + B - 1) / B, B, 0, stream>>>(h1, ei, w, dinv, acc1, E);
    gcn_finalize<<<(NF + B - 1) / B, B, 0, stream>>>(acc1, h1, dinv, b1, acc1, NF, 1);

    // ---- layer 2 ----
    gcn_matmul_wmma<16><<<mmGrid, B, 0, stream>>>(acc1, W2, h2, ntiles);
    fill_f32<<<(NF + B - 1) / B, B, 0, stream>>>(outf, 0.0f, NF);
    gcn_edge_scatter<<<(E + B - 1) / B, B, 0, stream>>>(h2, ei, w, dinv, outf, E);
    gcn_finalize<<<(NF + B - 1) / B, B, 0, stream>>>(outf, h2, dinv, b2, outf, NF, 0);
}